// GAT_18202071400537
// MI455X (gfx1250) — compile-verified
//
#include <hip/hip_runtime.h>
#include <hip/hip_bf16.h>

#define GAT_N_IN   128
#define GAT_H      4
#define GAT_C      32
#define NEG_SLOPE  0.2f

typedef __attribute__((ext_vector_type(2))) float v2f;
typedef __attribute__((ext_vector_type(8))) float v8f;

__device__ __forceinline__ float leaky(float v) {
    return v > 0.f ? v : NEG_SLOPE * v;
}

__device__ __forceinline__ void atomAddF(float* p, float v) {
    __hip_atomic_fetch_add(p, v, __ATOMIC_RELAXED, __HIP_MEMORY_SCOPE_AGENT);
}

// Monotonic order-preserving float<->uint encoding for atomicMax on floats.
__device__ __forceinline__ unsigned fkey(float v) {
    unsigned b = __float_as_uint(v);
    return (b & 0x80000000u) ? ~b : (b | 0x80000000u);
}
__device__ __forceinline__ float funkey(unsigned k) {
    return (k & 0x80000000u) ? __uint_as_float(k & 0x7FFFFFFFu)
                             : __uint_as_float(~k);
}

// ---------------------------------------------------------------------------
// Per-layer state init: out accumulator = 0, denom = 0, emax key = 0 (== -inf)
// ---------------------------------------------------------------------------
__global__ void gat_init(float* __restrict__ acc, float* __restrict__ denom,
                         unsigned* __restrict__ emax_u, int n) {
    int i = blockIdx.x * blockDim.x + threadIdx.x;
    if (i < n * GAT_N_IN) acc[i] = 0.f;
    if (i < n * GAT_H) { denom[i] = 0.f; emax_u[i] = 0u; }
}

// ---------------------------------------------------------------------------
// h = X[N,128] @ W[128,128] via V_WMMA_F32_16X16X4_F32.
// Block = 256 threads = 8 waves; wave w owns the 16x16 output tile at
// (blockIdx.x*16, w*16). K loop: 128/4 = 32 WMMAs.
// A 16x4 f32 layout: lane L holds row M=L&15; VGPR0 = K+(L<16?0:2), VGPR1 = +1.
// B 4x16 f32 layout: lane L holds col N=L&15; VGPR0 = row K+(L<16?0:2), VGPR1 = +1.
// C/D 16x16 f32: VGPR v holds row v (lanes 0-15) / v+8 (lanes 16-31), col = L&15.
// ---------------------------------------------------------------------------
__global__ __launch_bounds__(256) void gat_gemm_wmma(
    const float* __restrict__ X, const float* __restrict__ W,
    float* __restrict__ Hout) {
    const int wave = threadIdx.x >> 5;
    const int lane = threadIdx.x & 31;
    const int l15  = lane & 15;
    const bool hi  = lane >= 16;

    const int row  = blockIdx.x * 16 + l15;   // A row this lane feeds
    const int col  = wave * 16 + l15;         // B/C column this lane feeds

    const float* __restrict__ xrow = X + (size_t)row * GAT_N_IN;

    v8f acc = {};
#if __has_builtin(__builtin_amdgcn_wmma_f32_16x16x4_f32)
    #pragma unroll
    for (int k = 0; k < GAT_N_IN; k += 4) {
        const int ka = k + (hi ? 2 : 0);
        v2f a, b;
        a.x = xrow[ka];
        a.y = xrow[ka + 1];
        b.x = W[(size_t)ka * GAT_N_IN + col];
        b.y = W[(size_t)(ka + 1) * GAT_N_IN + col];
        acc = __builtin_amdgcn_wmma_f32_16x16x4_f32(
            /*neg_a=*/false, a, /*neg_b=*/false, b,
            /*c_mod=*/(short)0, acc, /*reuse_a=*/false, /*reuse_b=*/false);
    }
#else
    // Fallback (should not trigger on gfx1250): scalar FMA over this lane's tile.
    // Each lane computes rows (blockIdx.x*16 + v + hi*8), col = col.
    for (int v = 0; v < 8; ++v) {
        const float* xr = X + (size_t)(blockIdx.x * 16 + v + (hi ? 8 : 0)) * GAT_N_IN;
        float s = 0.f;
        for (int k = 0; k < GAT_N_IN; ++k) s = fmaf(xr[k], W[(size_t)k * GAT_N_IN + col], s);
        acc[v] = s;
    }
#endif
    const int r0 = blockIdx.x * 16 + (hi ? 8 : 0);
    #pragma unroll
    for (int v = 0; v < 8; ++v)
        Hout[(size_t)(r0 + v) * GAT_N_IN + col] = acc[v];
}

// ---------------------------------------------------------------------------
// alpha_src[n,h] = sum_c h[n,h,c]*a_src[h,c] ; same for alpha_dst. 1 thread per (n,h).
// ---------------------------------------------------------------------------
__global__ void gat_alpha(const float* __restrict__ Hm,
                          const float* __restrict__ a_s, const float* __restrict__ a_d,
                          float* __restrict__ as_out, float* __restrict__ ad_out, int n) {
    int i = blockIdx.x * blockDim.x + threadIdx.x;
    if (i >= n * GAT_H) return;
    const int head = i & (GAT_H - 1);
    const int node = i >> 2;
    const float* hr = Hm + (size_t)node * GAT_N_IN + head * GAT_C;
    const float* vs = a_s + head * GAT_C;
    const float* vd = a_d + head * GAT_C;
    float s1 = 0.f, s2 = 0.f;
    #pragma unroll
    for (int c = 0; c < GAT_C; ++c) {
        float v = hr[c];
        s1 = fmaf(v, vs[c], s1);
        s2 = fmaf(v, vd[c], s2);
    }
    as_out[i] = s1;
    ad_out[i] = s2;
}

__device__ __forceinline__ void edge_sd(const long long* __restrict__ ei, int E_, int e,
                                        int& s, int& d) {
    if (e < E_) { s = (int)ei[e]; d = (int)ei[E_ + e]; }
    else        { s = d = e - E_; }
}

// Pass 1: segment max of leaky_relu(alpha_src[src]+alpha_dst[dst]) over dst.
__global__ void gat_edge_max(const long long* __restrict__ ei, int E_, int n,
                             const float* __restrict__ as, const float* __restrict__ ad,
                             unsigned* __restrict__ emax_u) {
    int e = blockIdx.x * blockDim.x + threadIdx.x;
    if (e >= E_ + n) return;
    int s, d; edge_sd(ei, E_, e, s, d);
    #pragma unroll
    for (int h = 0; h < GAT_H; ++h) {
        float v = leaky(as[s * GAT_H + h] + ad[d * GAT_H + h]);
        atomicMax(&emax_u[d * GAT_H + h], fkey(v));
    }
}

// Pass 2: ex = exp(e - emax[dst]); store per-edge, atomicAdd into denom[dst].
__global__ void gat_edge_exp(const long long* __restrict__ ei, int E_, int n,
                             const float* __restrict__ as, const float* __restrict__ ad,
                             const unsigned* __restrict__ emax_u,
                             float* __restrict__ ex, float* __restrict__ denom) {
    int e = blockIdx.x * blockDim.x + threadIdx.x;
    if (e >= E_ + n) return;
    int s, d; edge_sd(ei, E_, e, s, d);
    #pragma unroll
    for (int h = 0; h < GAT_H; ++h) {
        float v = leaky(as[s * GAT_H + h] + ad[d * GAT_H + h]);
        float m = funkey(emax_u[d * GAT_H + h]);
        float x = expf(v - m);
        ex[(size_t)e * GAT_H + h] = x;
        atomAddF(&denom[d * GAT_H + h], x);
    }
}

// Pass 3: one wave per edge. lane covers channels [lane*4, lane*4+4),
// head = lane/8. out[dst] += h[src] * (ex/denom[dst]).
__global__ __launch_bounds__(256) void gat_edge_agg(
    const long long* __restrict__ ei, int E_, int n,
    const float* __restrict__ Hm, const float* __restrict__ ex,
    const float* __restrict__ denom, float* __restrict__ out) {
    int wid  = (blockIdx.x * blockDim.x + threadIdx.x) >> 5;
    int lane = threadIdx.x & 31;
    if (wid >= E_ + n) return;
    int s, d; edge_sd(ei, E_, wid, s, d);
    const int head = lane >> 3;           // 8 lanes per head (8*4 = 32 channels)
    const float w = ex[(size_t)wid * GAT_H + head] / denom[d * GAT_H + head];
    const float4 hv = *(const float4*)(Hm + (size_t)s * GAT_N_IN + lane * 4);
    float* o = out + (size_t)d * GAT_N_IN + lane * 4;
    atomAddF(o + 0, hv.x * w);
    atomAddF(o + 1, hv.y * w);
    atomAddF(o + 2, hv.z * w);
    atomAddF(o + 3, hv.w * w);
}

// Finalize: dst = (acc + bias), optional ReLU.
__global__ void gat_finalize(const float* __restrict__ acc, const float* __restrict__ bias,
                             float* __restrict__ dst, int n, int relu) {
    int i = blockIdx.x * blockDim.x + threadIdx.x;
    if (i >= n * GAT_N_IN) return;
    float v = acc[i] + bias[i & (GAT_N_IN - 1)];
    if (relu) v = fmaxf(v, 0.f);
    dst[i] = v;
}

extern "C" void kernel_launch(void* const* d_in, const int* in_sizes, int n_in,
                              void* d_out, int out_size, void* d_ws, size_t ws_size,
                              hipStream_t stream) {
    const float*      x  = (const float*)d_in[0];
    const long long*  ei = (const long long*)d_in[1];
    const int N  = in_sizes[0] / GAT_N_IN;   // 50000
    const int E_ = in_sizes[1] / 2;          // 800000
    const int total = E_ + N;                // edges incl. self loops

    // Workspace carve-up (256B aligned slices).
    char*  base = (char*)d_ws;
    size_t off  = 0;
    auto carve = [&](size_t bytes) {
        void* p = base + off;
        off = (off + bytes + 255) & ~(size_t)255;
        return p;
    };
    float*    bufA   = (float*)   carve((size_t)N * GAT_N_IN * sizeof(float));
    float*    bufB   = (float*)   carve((size_t)N * GAT_N_IN * sizeof(float));
    float*    hbuf   = (float*)   carve((size_t)N * GAT_N_IN * sizeof(float));
    float*    asb    = (float*)   carve((size_t)N * GAT_H * sizeof(float));
    float*    adb    = (float*)   carve((size_t)N * GAT_H * sizeof(float));
    unsigned* emax_u = (unsigned*)carve((size_t)N * GAT_H * sizeof(unsigned));
    float*    denom  = (float*)   carve((size_t)N * GAT_H * sizeof(float));
    float*    exbuf  = (float*)   carve((size_t)total * GAT_H * sizeof(float));
    (void)ws_size; (void)n_in; (void)out_size;

    const int TB = 256;
    const int gInit  = (N * GAT_N_IN + TB - 1) / TB;
    const int gGemm  = N / 16;                       // 50000/16 = 3125 exact
    const int gAlpha = (N * GAT_H + TB - 1) / TB;
    const int gEdge  = (total + TB - 1) / TB;
    const int gAgg   = (total + (TB / 32) - 1) / (TB / 32);   // wave per edge
    const int gFin   = gInit;

    const float* xin = x;
    for (int L = 0; L < 4; ++L) {
        const float* W    = (const float*)d_in[2 + 4 * L];
        const float* a_s  = (const float*)d_in[3 + 4 * L];
        const float* a_d  = (const float*)d_in[4 + 4 * L];
        const float* bias = (const float*)d_in[5 + 4 * L];
        float* acc = (L & 1) ? bufB : bufA;

        gat_init     <<<gInit,  TB, 0, stream>>>(acc, denom, emax_u, N);
        gat_gemm_wmma<<<gGemm,  TB, 0, stream>>>(xin, W, hbuf);
        gat_alpha    <<<gAlpha, TB, 0, stream>>>(hbuf, a_s, a_d, asb, adb, N);
        gat_edge_max <<<gEdge,  TB, 0, stream>>>(ei, E_, N, asb, adb, emax_u);
        gat_edge_exp <<<gEdge,  TB, 0, stream>>>(ei, E_, N, asb, adb, emax_u, exbuf, denom);
        gat_edge_agg <<<gAgg,   TB, 0, stream>>>(ei, E_, N, hbuf, exbuf, denom, acc);

        float* dst = (L == 3) ? (float*)d_out : acc;   // finalize in place or to d_out
        gat_finalize <<<gFin,   TB, 0, stream>>>(acc, bias, dst, N, L < 3 ? 1 : 0);
        xin = dst;
    }
}